// SphericalSamplingDecoder_51926154609190
// MI455X (gfx1250) — compile-verified
//
#include <hip/hip_runtime.h>
#include <math.h>

typedef __attribute__((ext_vector_type(2))) float v2f;
typedef __attribute__((ext_vector_type(4))) float v4f;
typedef __attribute__((ext_vector_type(8))) float v8f;

#define WAVES  4
#define BLOCK  (WAVES * 32)

static __device__ __forceinline__ v8f wmma_f32_k4(v2f a, v2f b, v8f c) {
    return __builtin_amdgcn_wmma_f32_16x16x4_f32(false, a, false, b,
                                                 (short)0, c, false, false);
}

// signal tile: C[16x16] = f4_tile(16x9) * Y4^T(9x16), K padded 9->12
static __device__ __forceinline__ v8f signal_tile(const float* __restrict__ yrow,
                                                  int half, const v2f a0,
                                                  const v2f a1, const v2f a2) {
    const float* ph = yrow + 2 * half;      // K-base for this lane half
    v2f b0, b1, b2;
    b0.x = ph[0]; b0.y = ph[1];             // K = {0,1} / {2,3}
    b1.x = ph[4]; b1.y = ph[5];             // K = {4,5} / {6,7}
    const float y8 = yrow[8];               // loaded by all lanes (branchless)
    b2.x = half ? 0.0f : y8;                // K = {8,0} / {0,0}
    b2.y = 0.0f;
    v8f c = {};
    c = wmma_f32_k4(a0, b0, c);
    c = wmma_f32_k4(a1, b1, c);
    c = wmma_f32_k4(a2, b2, c);
    return c;
}

// dots tile: D[16x16] = z_axis_tile(16x3) * grid^T(3x16), K padded 3->4
static __device__ __forceinline__ v8f dots_tile(const float* __restrict__ grow,
                                                int half, v2f az) {
    const float g0 = grow[0], g1 = grow[1], g2 = grow[2];
    v2f bg;
    bg.x = half ? g2 : g0;
    bg.y = half ? 0.0f : g1;
    v8f d = {};
    return wmma_f32_k4(az, bg, d);
}

static __device__ __forceinline__ void half16_reduce_max(float& val, int& idx) {
    // reduce (val,idx) max over each 16-lane half of the wave32;
    // xor masks 1..8 never cross the half boundary
    #pragma unroll
    for (int m = 8; m >= 1; m >>= 1) {
        float ov = __shfl_xor(val, m, 32);
        int   oi = __shfl_xor(idx, m, 32);
        if (ov > val || (ov == val && oi < idx)) { val = ov; idx = oi; }
    }
}

__global__ __launch_bounds__(BLOCK)
void sph_decoder_kernel(const float* __restrict__ f4,
                        const float* __restrict__ grid,
                        const float* __restrict__ Y4,
                        float* __restrict__ out,
                        int n_fib)
{
    __shared__ float s_val[WAVES * 16];
    __shared__ int   s_idx[WAVES * 16];
    __shared__ float s_z[16][3];

    const int tid  = threadIdx.x;
    const int lane = tid & 31;
    const int wave = tid >> 5;
    const int half = lane >> 4;   // 0: K0/K1 lanes, 1: K2/K3 lanes (A/B frag layout)
    const int l16  = lane & 15;
    const int rowbase = blockIdx.x * 16;

    const int ntiles_full = n_fib >> 4;           // fully populated N-tiles
    const int tail        = n_fib & 15;           // 0 for n_fib = 10000

    // ---- A fragments: f4 tile, 16 rows x K=9 padded to 12 (3 chunks of K=4) ----
    v2f a0, a1, a2;
    {
        const float* frow = f4 + (size_t)(rowbase + l16) * 9;
        const float* pf = frow + 2 * half;
        a0.x = pf[0]; a0.y = pf[1];
        a1.x = pf[4]; a1.y = pf[5];
        const float f8 = frow[8];
        a2.x = half ? 0.0f : f8;
        a2.y = 0.0f;
    }

    // ================= Phase 1: argmax_g signal[b,g] =================
    float bv[8]; int bi[8];
    #pragma unroll
    for (int v = 0; v < 8; ++v) { bv[v] = -INFINITY; bi[v] = 0; }

    for (int nt = wave; nt < ntiles_full; nt += WAVES) {
        const int g = nt * 16 + l16;
        const v8f c = signal_tile(Y4 + (size_t)g * 9, half, a0, a1, a2);
        #pragma unroll
        for (int v = 0; v < 8; ++v) {
            const bool gt = c[v] > bv[v];
            bv[v] = gt ? c[v] : bv[v];
            bi[v] = gt ? g    : bi[v];
        }
    }
    if (tail && ((ntiles_full % WAVES) == wave)) {  // partial last tile (uniform per wave)
        const int g  = ntiles_full * 16 + l16;
        const int gc = (g < n_fib) ? g : (n_fib - 1);
        v8f c = signal_tile(Y4 + (size_t)gc * 9, half, a0, a1, a2);
        #pragma unroll
        for (int v = 0; v < 8; ++v) {
            const float cv = (g < n_fib) ? c[v] : -INFINITY;
            const bool gt = cv > bv[v];
            bv[v] = gt ? cv : bv[v];
            bi[v] = gt ? g  : bi[v];
        }
    }

    #pragma unroll
    for (int v = 0; v < 8; ++v) half16_reduce_max(bv[v], bi[v]);
    if (l16 == 0) {  // lane 0 -> rows 0..7, lane 16 -> rows 8..15
        #pragma unroll
        for (int v = 0; v < 8; ++v) {
            s_val[wave * 16 + half * 8 + v] = bv[v];
            s_idx[wave * 16 + half * 8 + v] = bi[v];
        }
    }
    __syncthreads();

    if (tid < 16) {
        float vv = s_val[tid]; int gg = s_idx[tid];
        for (int w = 1; w < WAVES; ++w) {
            const float ov = s_val[w * 16 + tid];
            const int   og = s_idx[w * 16 + tid];
            if (ov > vv || (ov == vv && og < gg)) { vv = ov; gg = og; }
        }
        s_z[tid][0] = grid[(size_t)gg * 3 + 0];
        s_z[tid][1] = grid[(size_t)gg * 3 + 1];
        s_z[tid][2] = grid[(size_t)gg * 3 + 2];
    }
    __syncthreads();

    // ---- A fragment for dots GEMM: z_axis 16 rows x K=3 padded to 4 ----
    v2f az;
    az.x = half ? s_z[l16][2] : s_z[l16][0];
    az.y = half ? 0.0f        : s_z[l16][1];

    // ============ Phase 2: argmax_g of masked signal ============
    #pragma unroll
    for (int v = 0; v < 8; ++v) { bv[v] = -INFINITY; bi[v] = 0; }

    for (int nt = wave; nt < ntiles_full; nt += WAVES) {
        const int g = nt * 16 + l16;
        const v8f c = signal_tile(Y4 + (size_t)g * 9, half, a0, a1, a2);
        const v8f d = dots_tile(grid + (size_t)g * 3, half, az);
        #pragma unroll
        for (int v = 0; v < 8; ++v) {
            const float val = (fabsf(d[v]) < 0.2f) ? c[v] : -INFINITY;
            const bool gt = val > bv[v];
            bv[v] = gt ? val : bv[v];
            bi[v] = gt ? g   : bi[v];
        }
    }
    if (tail && ((ntiles_full % WAVES) == wave)) {
        const int g  = ntiles_full * 16 + l16;
        const int gc = (g < n_fib) ? g : (n_fib - 1);
        const v8f c = signal_tile(Y4 + (size_t)gc * 9, half, a0, a1, a2);
        const v8f d = dots_tile(grid + (size_t)gc * 3, half, az);
        #pragma unroll
        for (int v = 0; v < 8; ++v) {
            float val = (fabsf(d[v]) < 0.2f) ? c[v] : -INFINITY;
            val = (g < n_fib) ? val : -INFINITY;
            const bool gt = val > bv[v];
            bv[v] = gt ? val : bv[v];
            bi[v] = gt ? g   : bi[v];
        }
    }

    #pragma unroll
    for (int v = 0; v < 8; ++v) half16_reduce_max(bv[v], bi[v]);
    if (l16 == 0) {
        #pragma unroll
        for (int v = 0; v < 8; ++v) {
            s_val[wave * 16 + half * 8 + v] = bv[v];
            s_idx[wave * 16 + half * 8 + v] = bi[v];
        }
    }
    __syncthreads();

    // ============ Final: Gram-Schmidt + quaternion, one thread per row ============
    if (tid < 16) {
        float vv = s_val[tid]; int gg = s_idx[tid];
        for (int w = 1; w < WAVES; ++w) {
            const float ov = s_val[w * 16 + tid];
            const int   og = s_idx[w * 16 + tid];
            if (ov > vv || (ov == vv && og < gg)) { vv = ov; gg = og; }
        }
        const float xa0 = grid[(size_t)gg * 3 + 0];
        const float xa1 = grid[(size_t)gg * 3 + 1];
        const float xa2 = grid[(size_t)gg * 3 + 2];

        float zx = s_z[tid][0], zy = s_z[tid][1], zz = s_z[tid][2];
        const float zn = fmaxf(sqrtf(zx*zx + zy*zy + zz*zz), 1e-12f);
        zx /= zn; zy /= zn; zz /= zn;

        const float pr = xa0*zx + xa1*zy + xa2*zz;
        float x0 = xa0 - pr*zx, x1 = xa1 - pr*zy, x2 = xa2 - pr*zz;
        const float xn = fmaxf(sqrtf(x0*x0 + x1*x1 + x2*x2), 1e-12f);
        x0 /= xn; x1 /= xn; x2 /= xn;

        const float y0 = zy*x2 - zz*x1;
        const float y1 = zz*x0 - zx*x2;
        const float y2 = zx*x1 - zy*x0;

        // R columns are (x, y, z)
        const float m00 = x0, m01 = y0, m02 = zx;
        const float m10 = x1, m11 = y1, m12 = zy;
        const float m20 = x2, m21 = y2, m22 = zz;

        const float q0 = sqrtf(fmaxf(1.0f + m00 + m11 + m22, 0.0f));
        const float q1 = sqrtf(fmaxf(1.0f + m00 - m11 - m22, 0.0f));
        const float q2 = sqrtf(fmaxf(1.0f - m00 + m11 - m22, 0.0f));
        const float q3 = sqrtf(fmaxf(1.0f - m00 - m11 + m22, 0.0f));

        // argmax over q (first occurrence on ties, matching numpy)
        const bool b1q = q1 > q0;
        float bq = b1q ? q1 : q0;
        const bool b2q = q2 > bq;  bq = b2q ? q2 : bq;
        const bool b3q = q3 > bq;  bq = b3q ? q3 : bq;

        // candidate rows (scalar selects -> v_cndmask, no scratch)
        float c0 = q0*q0,     c1 = m21 - m12, c2 = m02 - m20, c3 = m10 - m01;
        c0 = b1q ? (m21 - m12) : c0;
        c1 = b1q ? (q1*q1)     : c1;
        c2 = b1q ? (m10 + m01) : c2;
        c3 = b1q ? (m02 + m20) : c3;
        c0 = b2q ? (m02 - m20) : c0;
        c1 = b2q ? (m10 + m01) : c1;
        c2 = b2q ? (q2*q2)     : c2;
        c3 = b2q ? (m12 + m21) : c3;
        c0 = b3q ? (m10 - m01) : c0;
        c1 = b3q ? (m20 + m02) : c1;
        c2 = b3q ? (m21 + m12) : c2;
        c3 = b3q ? (q3*q3)     : c3;

        const float inv = 1.0f / (2.0f * fmaxf(bq, 0.1f));
        v4f q; q.x = c0 * inv; q.y = c1 * inv; q.z = c2 * inv; q.w = c3 * inv;
        *(v4f*)(out + (size_t)(rowbase + tid) * 4) = q;  // 16B-aligned store
    }
}

extern "C" void kernel_launch(void* const* d_in, const int* in_sizes, int n_in,
                              void* d_out, int out_size, void* d_ws, size_t ws_size,
                              hipStream_t stream) {
    (void)n_in; (void)d_ws; (void)ws_size; (void)out_size;
    const float* f4   = (const float*)d_in[0];
    // d_in[1] (f6) is unused by the reference
    const float* grid = (const float*)d_in[2];
    const float* Y4   = (const float*)d_in[3];
    float* out = (float*)d_out;

    const int batch   = in_sizes[0] / 9;   // 8192
    const int n_fib   = in_sizes[2] / 3;   // 10000
    const int nblocks = batch / 16;        // 512

    sph_decoder_kernel<<<nblocks, BLOCK, 0, stream>>>(f4, grid, Y4, out, n_fib);
}